// ConLID_H_60284160966936
// MI455X (gfx1250) — compile-verified
//
#include <hip/hip_runtime.h>
#include <hip/hip_bf16.h>
#include <math.h>

typedef __attribute__((ext_vector_type(2))) float v2f;
typedef __attribute__((ext_vector_type(8))) float v8f;

namespace {
constexpr int kB = 1024;   // batch
constexpr int kL = 256;    // seq len
constexpr int kE = 256;    // embed dim
constexpr int kC = 2000;   // classes
constexpr float kInvTemp = 10.0f;  // 1 / 0.1
constexpr int kMinNeg = 10;
}

// ---------------- accumulator init (harness does not re-zero ws) -----------
__global__ void init_accum_kernel(float* __restrict__ accum) {
  if (threadIdx.x < 3) accum[threadIdx.x] = 0.0f;
}

// ---------------- masked mean pooling --------------------------------------
// 1 block per batch row; 256 threads = one embedding lane each.
__global__ void pool_kernel(const int* __restrict__ ids,
                            const float* __restrict__ emb,
                            float* __restrict__ pooled) {
  const int b = blockIdx.x;
  const int e = threadIdx.x;
  __shared__ int s_ids[kL];
  s_ids[e] = ids[b * kL + e];
  __syncthreads();
  float acc = 0.0f;
  float cnt = 0.0f;
  for (int l = 0; l < kL; ++l) {
    const int id = s_ids[l];
    if (id != 0 && id != 1) {  // not UNK, not PAD (uniform branch per block)
      acc += emb[(size_t)id * kE + e];
      cnt += 1.0f;
    }
  }
  pooled[(size_t)b * kE + e] = acc / cnt;
}

// ---------------- logits = pooled @ fc_w + fc_b  (fp32 WMMA) ---------------
// One wave computes a 16x16 output tile; K-loop of 64 x V_WMMA_F32_16X16X4_F32.
// fp32 A 16x4 layout: lanes 0-15 -> M=lane, K={k0,k0+1}; lanes 16-31 -> K={k0+2,k0+3}.
// fp32 B 4x16 layout mirrors it with N=lane%16.
// C/D: VGPR r holds row m0+r (lanes 0-15) and m0+r+8 (lanes 16-31), col n0+lane%16.
__global__ void logits_wmma_kernel(const float* __restrict__ pooled,
                                   const float* __restrict__ fcw,  // [E][C]
                                   const float* __restrict__ fcb,
                                   float* __restrict__ logits) {
  const int wave = blockIdx.x * (blockDim.x >> 5) + (threadIdx.x >> 5);
  const int lane = threadIdx.x & 31;
  const int tn = wave % (kC / 16);
  const int tm = wave / (kC / 16);
  const int m0 = tm * 16;
  const int n0 = tn * 16;
  const int half = lane >> 4;
  const int lid = lane & 15;

  v8f acc = {};
  const float* aptr = pooled + (size_t)(m0 + lid) * kE + half * 2;
  const float* bptr = fcw + (size_t)(half * 2) * kC + (n0 + lid);
#pragma unroll 4
  for (int k0 = 0; k0 < kE; k0 += 4) {
    v2f a;  a.x = aptr[0];  a.y = aptr[1];
    v2f bv; bv.x = bptr[0]; bv.y = bptr[kC];
    acc = __builtin_amdgcn_wmma_f32_16x16x4_f32(false, a, false, bv,
                                                (short)0, acc, false, false);
    aptr += 4;
    bptr += 4 * (size_t)kC;
  }
  const float bias = fcb[n0 + lid];
  const int row0 = m0 + half * 8;
  const int col = n0 + lid;
#pragma unroll
  for (int r = 0; r < 8; ++r) {
    logits[(size_t)(row0 + r) * kC + col] = acc[r] + bias;
  }
}

// ---------------- cross-entropy --------------------------------------------
__global__ void ce_kernel(const float* __restrict__ logits,
                          const int* __restrict__ labels,
                          float* __restrict__ accum) {
  const int b = blockIdx.x;
  const int t = threadIdx.x;  // 256
  __shared__ float red[256];
  const float* row = logits + (size_t)b * kC;
  float m = -3.4e38f;
  for (int c = t; c < kC; c += 256) m = fmaxf(m, row[c]);
  red[t] = m;
  __syncthreads();
  for (int s = 128; s > 0; s >>= 1) {
    if (t < s) red[t] = fmaxf(red[t], red[t + s]);
    __syncthreads();
  }
  m = red[0];
  __syncthreads();
  float sum = 0.0f;
  for (int c = t; c < kC; c += 256) sum += expf(row[c] - m);
  red[t] = sum;
  __syncthreads();
  for (int s = 128; s > 0; s >>= 1) {
    if (t < s) red[t] += red[t + s];
    __syncthreads();
  }
  if (t == 0) {
    const float lse = m + logf(red[0]);
    const float lp = row[labels[b]] - lse;
    atomicAdd(&accum[0], -lp * (1.0f / (float)kB));
  }
}

// ---------------- L2 normalize ---------------------------------------------
__global__ void norm_kernel(const float* __restrict__ pooled,
                            float* __restrict__ nemb) {
  const int b = blockIdx.x;
  const int t = threadIdx.x;  // 256 == kE
  __shared__ float red[kE];
  const float v = pooled[(size_t)b * kE + t];
  red[t] = v * v;
  __syncthreads();
  for (int s = kE / 2; s > 0; s >>= 1) {
    if (t < s) red[t] += red[t + s];
    __syncthreads();
  }
  nemb[(size_t)b * kE + t] = v * (1.0f / sqrtf(red[0]));
}

// ---------------- sim = nemb @ nemb^T / TEMP  (fp32 WMMA) ------------------
__global__ void sim_wmma_kernel(const float* __restrict__ nemb,
                                float* __restrict__ sim) {
  const int wave = blockIdx.x * (blockDim.x >> 5) + (threadIdx.x >> 5);
  const int lane = threadIdx.x & 31;
  const int tn = wave % (kB / 16);
  const int tm = wave / (kB / 16);
  const int m0 = tm * 16;
  const int n0 = tn * 16;
  const int half = lane >> 4;
  const int lid = lane & 15;

  v8f acc = {};
  const float* aptr = nemb + (size_t)(m0 + lid) * kE + half * 2;
  const float* bptr = nemb + (size_t)(n0 + lid) * kE + half * 2;  // B = nemb^T
#pragma unroll 4
  for (int k0 = 0; k0 < kE; k0 += 4) {
    v2f a;  a.x = aptr[0];  a.y = aptr[1];
    v2f bv; bv.x = bptr[0]; bv.y = bptr[1];
    acc = __builtin_amdgcn_wmma_f32_16x16x4_f32(false, a, false, bv,
                                                (short)0, acc, false, false);
    aptr += 4;
    bptr += 4;
  }
  const int row0 = m0 + half * 8;
  const int col = n0 + lid;
#pragma unroll
  for (int r = 0; r < 8; ++r) {
    sim[(size_t)(row0 + r) * kB + col] = acc[r] * kInvTemp;
  }
}

// ---------------- contrastive loss (per row) --------------------------------
__global__ void con_kernel(const float* __restrict__ sim,
                           const int* __restrict__ labels,
                           const int* __restrict__ doc,
                           const int* __restrict__ dom,
                           const int* __restrict__ domg,
                           const int* __restrict__ scr,
                           float* __restrict__ accum) {
  const int i = blockIdx.x;
  const int t = threadIdx.x;  // 256
  __shared__ int s_cnt[8];
  __shared__ float s_red[256];
  __shared__ int s_idx;
  if (t < 8) s_cnt[t] = 0;
  __syncthreads();

  const int li = labels[i], di = doc[i], mi = dom[i], gi = domg[i], si = scr[i];

  int c0 = 0, c1 = 0, c2 = 0, c3 = 0, c4 = 0, c5 = 0, c6 = 0, c7 = 0;
  for (int j = t; j < kB; j += 256) {
    const bool neq = labels[j] != li;
    const bool sc = scr[j] == si;
    const bool dc = doc[j] == di;
    const bool dm = dom[j] == mi;
    const bool dg = domg[j] == gi;
    c0 += (neq && sc && dc) ? 1 : 0;
    c1 += (neq && sc && dm) ? 1 : 0;
    c2 += (neq && sc && dg) ? 1 : 0;
    c3 += (neq && sc) ? 1 : 0;
    c4 += (neq && dc) ? 1 : 0;
    c5 += (neq && dm) ? 1 : 0;
    c6 += (neq && dg) ? 1 : 0;
    c7 += neq ? 1 : 0;
  }
  atomicAdd(&s_cnt[0], c0); atomicAdd(&s_cnt[1], c1);
  atomicAdd(&s_cnt[2], c2); atomicAdd(&s_cnt[3], c3);
  atomicAdd(&s_cnt[4], c4); atomicAdd(&s_cnt[5], c5);
  atomicAdd(&s_cnt[6], c6); atomicAdd(&s_cnt[7], c7);
  __syncthreads();

  if (t == 0) {
    int idx = 7;  // fallback: last condition
    for (int k = 7; k >= 0; --k)
      if (s_cnt[k] >= kMinNeg) idx = k;  // ends at first (lowest) meeting cond
    s_idx = idx;
  }
  __syncthreads();
  const int idx = s_idx;

  float pos = 0.0f, neg = 0.0f;
  for (int j = t; j < kB; j += 256) {
    const float es = expf(sim[(size_t)i * kB + j]);
    const bool same = labels[j] == li;
    if (same && j != i) pos += es;
    const bool neq = !same;
    const bool sc = scr[j] == si;
    const bool dc = doc[j] == di;
    const bool dm = dom[j] == mi;
    const bool dg = domg[j] == gi;
    bool sel;
    switch (idx) {
      case 0:  sel = neq && sc && dc; break;
      case 1:  sel = neq && sc && dm; break;
      case 2:  sel = neq && sc && dg; break;
      case 3:  sel = neq && sc;       break;
      case 4:  sel = neq && dc;       break;
      case 5:  sel = neq && dm;       break;
      case 6:  sel = neq && dg;       break;
      default: sel = neq;             break;
    }
    if (sel) neg += es;
  }

  s_red[t] = pos;
  __syncthreads();
  for (int s = 128; s > 0; s >>= 1) {
    if (t < s) s_red[t] += s_red[t + s];
    __syncthreads();
  }
  const float pos_sum = s_red[0];
  __syncthreads();
  s_red[t] = neg;
  __syncthreads();
  for (int s = 128; s > 0; s >>= 1) {
    if (t < s) s_red[t] += s_red[t + s];
    __syncthreads();
  }
  const float neg_sum = s_red[0];

  if (t == 0 && pos_sum != 0.0f) {
    atomicAdd(&accum[1], -logf(pos_sum / (pos_sum + neg_sum)));
    atomicAdd(&accum[2], 1.0f);
  }
}

// ---------------- finalize --------------------------------------------------
__global__ void finalize_kernel(const float* __restrict__ accum,
                                float* __restrict__ out0) {
  if (threadIdx.x == 0) {
    const float cnt = accum[2];
    const float con = (cnt > 0.0f) ? accum[1] / cnt : 0.0f;
    out0[0] = accum[0] + con;
  }
}

extern "C" void kernel_launch(void* const* d_in, const int* in_sizes, int n_in,
                              void* d_out, int out_size, void* d_ws, size_t ws_size,
                              hipStream_t stream) {
  const int* input_ids = (const int*)d_in[0];
  const int* labels = (const int*)d_in[1];
  const int* doc_ids = (const int*)d_in[2];
  const int* dom_ids = (const int*)d_in[3];
  const int* dom_gen_ids = (const int*)d_in[4];
  const int* script_ids = (const int*)d_in[5];
  const float* emb_table = (const float*)d_in[6];
  const float* fc_w = (const float*)d_in[7];
  const float* fc_b = (const float*)d_in[8];

  float* out = (float*)d_out;      // [0] = loss, [1..] = logits (B*C)
  float* logits = out + 1;

  float* ws = (float*)d_ws;
  float* pooled = ws;                          // kB*kE floats (1 MB)
  float* nemb = pooled + (size_t)kB * kE;      // kB*kE floats (1 MB)
  float* sim = nemb + (size_t)kB * kE;         // kB*kB floats (4 MB)
  float* accum = sim + (size_t)kB * kB;        // 3 floats: ce, con_sum, con_cnt

  init_accum_kernel<<<1, 32, 0, stream>>>(accum);

  pool_kernel<<<kB, kE, 0, stream>>>(input_ids, emb_table, pooled);

  {  // 64 x 125 = 8000 wave-tiles, 4 waves/block
    const int waves = (kB / 16) * (kC / 16);
    logits_wmma_kernel<<<waves / 4, 128, 0, stream>>>(pooled, fc_w, fc_b, logits);
  }

  ce_kernel<<<kB, 256, 0, stream>>>(logits, labels, accum);

  norm_kernel<<<kB, kE, 0, stream>>>(pooled, nemb);

  {  // 64 x 64 = 4096 wave-tiles, 4 waves/block
    const int waves = (kB / 16) * (kB / 16);
    sim_wmma_kernel<<<waves / 4, 128, 0, stream>>>(nemb, sim);
  }

  con_kernel<<<kB, 256, 0, stream>>>(sim, labels, doc_ids, dom_ids,
                                     dom_gen_ids, script_ids, accum);

  finalize_kernel<<<1, 32, 0, stream>>>(accum, out);
}